// Basis_13460427506086
// MI455X (gfx1250) — compile-verified
//
#include <hip/hip_runtime.h>

typedef __attribute__((ext_vector_type(2))) float v2f;
typedef __attribute__((ext_vector_type(8))) float v8f;

#define LSTR 80   // LDS row stride (floats): 2*LSTR % 64 == 32 -> lane-halves hit disjoint bank groups

// ---------------------------------------------------------------------------
// Kernel 1: per-workgroup partial G = Phi^T Phi over a grid-stride row range.
// 16 waves per WG; wave w = (ti,tj) accumulates the 16x16 tile G[ti,tj].
// WMMA: D(16x16) += A(16x4: Phi^T slice) * B(4x16: Phi slice), fp32.
// ---------------------------------------------------------------------------
__global__ __launch_bounds__(512) void gram_partial(const float* __restrict__ Phi,
                                                    float* __restrict__ part,
                                                    int m, int nwg) {
    __shared__ float tile[64 * LSTR];
    const int tid  = threadIdx.x;
    const int lane = tid & 31;
    const int wave = tid >> 5;       // 0..15
    const int ti   = wave >> 2;      // tile row of G
    const int tj   = wave & 3;       // tile col of G
    const int li   = lane & 15;      // 0..15
    const int kh   = lane >> 4;      // lane half -> K pair {0,1} or {2,3}

    v8f acc = {};

    for (long rowbase = (long)blockIdx.x * 64; rowbase < (long)m; rowbase += (long)nwg * 64) {
        // Stage 64x64 block of Phi into LDS (each thread: 8 floats, coalesced)
        {
            const int r = tid >> 3;            // 0..63
            const int c = (tid & 7) * 8;       // 0,8,...,56
            const long row = rowbase + r;
            float* dst = &tile[r * LSTR + c];
            if (row < (long)m) {
                const float4* src = reinterpret_cast<const float4*>(Phi + row * 64 + c);
                float4 a = src[0], b = src[1];
                dst[0]=a.x; dst[1]=a.y; dst[2]=a.z; dst[3]=a.w;
                dst[4]=b.x; dst[5]=b.y; dst[6]=b.z; dst[7]=b.w;
            } else {
                #pragma unroll
                for (int q = 0; q < 8; ++q) dst[q] = 0.0f;   // K-padding must be zero
            }
        }
        __syncthreads();

        #pragma unroll
        for (int k0 = 0; k0 < 64; k0 += 4) {
            const int ka = k0 + 2 * kh;
            v2f va, vb;
            // A (Phi^T) tile: element (M=li, K) = Phi[rowbase+K][ti*16+li]
            va.x = tile[ka       * LSTR + ti * 16 + li];
            va.y = tile[(ka + 1) * LSTR + ti * 16 + li];
            // B (Phi) tile: element (K, N=li) = Phi[rowbase+K][tj*16+li]
            vb.x = tile[ka       * LSTR + tj * 16 + li];
            vb.y = tile[(ka + 1) * LSTR + tj * 16 + li];
            acc = __builtin_amdgcn_wmma_f32_16x16x4_f32(false, va, false, vb,
                                                        (short)0, acc, false, false);
        }
        __syncthreads();
    }

    // Write this WG's partial G tile. D layout: VGPR v, lane: M = v + 8*kh, N = li.
    float* p = part + (size_t)blockIdx.x * 4096;
    #pragma unroll
    for (int v = 0; v < 8; ++v) {
        const int gi = ti * 16 + v + 8 * kh;
        const int gj = tj * 16 + li;
        p[gi * 64 + gj] = acc[v];
    }
}

// ---------------------------------------------------------------------------
// Kernel 1b: reduce partial Gs -> G (4096 elements)
// ---------------------------------------------------------------------------
__global__ __launch_bounds__(256) void reduce_g(const float* __restrict__ part,
                                                float* __restrict__ G, int nwg) {
    const int e = blockIdx.x * 256 + threadIdx.x;   // 0..4095 (16 blocks x 256)
    float s = 0.0f;
    for (int w = 0; w < nwg; ++w) s += part[(size_t)w * 4096 + e];
    G[e] = s;
}

// ---------------------------------------------------------------------------
// Kernel 2: Cholesky G = U^T U (U upper), then Rinv = U^{-1}. One WG, 64 thr.
// ---------------------------------------------------------------------------
__global__ __launch_bounds__(64) void chol_inv(const float* __restrict__ G,
                                               float* __restrict__ Rinv) {
    __shared__ float A[64 * 64];
    __shared__ float X[64 * 64];
    const int t = threadIdx.x;   // 0..63

    for (int i = 0; i < 64; ++i) A[i * 64 + t] = G[i * 64 + t];
    __syncthreads();

    // Right-looking Cholesky producing U in the upper triangle of A.
    for (int k = 0; k < 64; ++k) {
        const float dk = sqrtf(A[k * 64 + k]);
        if (t >= k) A[k * 64 + t] = (t == k) ? dk : (A[k * 64 + t] / dk);
        __syncthreads();
        if (t > k) {
            const float uki = A[k * 64 + t];             // U[k][t]
            for (int j = t; j < 64; ++j)
                A[t * 64 + j] -= uki * A[k * 64 + j];    // rank-1 update
        }
        __syncthreads();
    }

    // Back-substitution: thread t owns column t of X = U^{-1} (upper tri).
    for (int i = 0; i < 64; ++i) X[i * 64 + t] = 0.0f;
    X[t * 64 + t] = 1.0f / A[t * 64 + t];
    for (int i = t - 1; i >= 0; --i) {
        float s = 0.0f;
        for (int k = i + 1; k <= t; ++k) s += A[i * 64 + k] * X[k * 64 + t];
        X[i * 64 + t] = -s / A[i * 64 + i];
    }
    __syncthreads();
    for (int i = 0; i < 64; ++i) Rinv[i * 64 + t] = X[i * 64 + t];
}

// ---------------------------------------------------------------------------
// Kernel 3: Q = Phi * Rinv. One WG per 64 rows; Rinv staged in LDS.
// Wave (ti,tj) computes output tile rows [ti*16,+16) x cols [tj*16,+16).
// ---------------------------------------------------------------------------
__global__ __launch_bounds__(512) void apply_rinv(const float* __restrict__ Phi,
                                                  const float* __restrict__ Rinv,
                                                  float* __restrict__ Q, int m) {
    __shared__ float B[64 * LSTR];
    const int tid  = threadIdx.x;
    const int lane = tid & 31;
    const int wave = tid >> 5;
    const int ti   = wave >> 2;
    const int tj   = wave & 3;
    const int li   = lane & 15;
    const int kh   = lane >> 4;

    // Stage Rinv (16 KB) into LDS
    {
        const int r = tid >> 3;
        const int c = (tid & 7) * 8;
        const float* src = Rinv + r * 64 + c;
        float* dst = &B[r * LSTR + c];
        #pragma unroll
        for (int q = 0; q < 8; ++q) dst[q] = src[q];
    }
    __syncthreads();

    const long rowbase = (long)blockIdx.x * 64;
    long arow = rowbase + ti * 16 + li;
    if (arow >= (long)m) arow = (long)m - 1;        // clamp loads; stores are masked
    const float2* aptr = reinterpret_cast<const float2*>(Phi + arow * 64);

    v8f acc = {};
    #pragma unroll
    for (int k0 = 0; k0 < 64; k0 += 4) {
        const int ka = k0 + 2 * kh;
        v2f va, vb;
        const float2 a = aptr[ka >> 1];              // Phi[arow][ka], Phi[arow][ka+1]
        va.x = a.x; va.y = a.y;
        vb.x = B[ka       * LSTR + tj * 16 + li];    // Rinv[ka][tj*16+li]
        vb.y = B[(ka + 1) * LSTR + tj * 16 + li];
        acc = __builtin_amdgcn_wmma_f32_16x16x4_f32(false, va, false, vb,
                                                    (short)0, acc, false, false);
    }

    #pragma unroll
    for (int v = 0; v < 8; ++v) {
        const long row = rowbase + ti * 16 + v + 8 * kh;
        if (row < (long)m) Q[row * 64 + tj * 16 + li] = acc[v];
    }
}

// ---------------------------------------------------------------------------
extern "C" void kernel_launch(void* const* d_in, const int* in_sizes, int n_in,
                              void* d_out, int out_size, void* d_ws, size_t ws_size,
                              hipStream_t stream) {
    const float* Phi = (const float*)d_in[0];
    float* out = (float*)d_out;
    const int m = in_sizes[0] / 64;            // 500000

    float* ws   = (float*)d_ws;
    float* G    = ws;                          // 4096 floats
    float* Rinv = ws + 4096;                   // 4096 floats
    float* part = ws + 8192;                   // nwg * 4096 floats

    long avail = ((long)(ws_size / 4) - 8192) / 4096;
    int nwg = (int)(avail < 1 ? 1 : (avail > 512 ? 512 : avail));

    gram_partial<<<nwg, 512, 0, stream>>>(Phi, part, m, nwg);
    reduce_g<<<16, 256, 0, stream>>>(part, G, nwg);
    chol_inv<<<1, 64, 0, stream>>>(G, Rinv);

    const int nblk = (m + 63) / 64;
    apply_rinv<<<nblk, 512, 0, stream>>>(Phi, Rinv, out, m);
}